// LoopConv2d_21251498180724
// MI455X (gfx1250) — compile-verified
//
#include <hip/hip_runtime.h>

typedef __attribute__((ext_vector_type(2))) float v2f;
typedef __attribute__((ext_vector_type(4))) float v4f;
typedef __attribute__((ext_vector_type(8))) float v8f;
typedef __attribute__((ext_vector_type(4))) int   v4i;

// v4i with explicit address spaces for the async-copy builtin
typedef __attribute__((address_space(1))) v4i gv4i;  // global
typedef __attribute__((address_space(3))) v4i lv4i;  // LDS

#define C_IN   64
#define C_OUT  128
#define HW     112
#define WCHUNK 9216          // 32 pairs * 288 floats: one (kh,kw) weight panel (36 KB)

// ---------------------------------------------------------------------------
// Async global->LDS helpers (CDNA5 GLOBAL_LOAD_ASYNC_TO_LDS_B128, ASYNCcnt)
// ---------------------------------------------------------------------------
__device__ __forceinline__ void copy16_g2l(const float* g, float* l) {
#if __has_builtin(__builtin_amdgcn_global_load_async_to_lds_b128)
    __builtin_amdgcn_global_load_async_to_lds_b128(
        (gv4i*)g, (lv4i*)l, /*offset=*/0, /*cpol=*/0);
#else
    *(v4f*)l = *(const v4f*)g;
#endif
}

__device__ __forceinline__ void wait_async0() {
#if __has_builtin(__builtin_amdgcn_s_wait_asynccnt)
    __builtin_amdgcn_s_wait_asynccnt(0);
#elif __has_builtin(__builtin_amdgcn_global_load_async_to_lds_b128)
    asm volatile("s_wait_asynccnt 0" ::: "memory");
#endif
}

// ---------------------------------------------------------------------------
// Repack OIHW weights into the LDS fragment image, per (kh,kw) chunk:
//   wrep[kpos*9216 + (ci>>1)*288 + co*2 + (ci&1)] = w[(co*C_IN+ci)*9 + kpos]
// 324 KB total, L2-resident; lets the main kernel stage W with contiguous
// async B128 copies instead of stride-9 gathers.
// ---------------------------------------------------------------------------
__global__ __launch_bounds__(256)
void repack_w(const float* __restrict__ w, float* __restrict__ wrep)
{
    const int idx  = blockIdx.x * 256 + threadIdx.x;   // 0 .. 9*8192-1
    const int kpos = idx >> 13;
    const int r    = idx & 8191;
    const int ci   = r >> 7;
    const int co   = r & 127;
    const float v  = w[(co * C_IN + ci) * 9 + kpos];
    wrep[kpos * WCHUNK + (ci >> 1) * 288 + co * 2 + (ci & 1)] = v;
}

// ---------------------------------------------------------------------------
// Pipelined main kernel: 128(co) x 16(spatial) tile per 256-thread block.
// LDS ping-pong: async W copies + register-staged im2col X for iteration k+1
// issue before the 16 WMMAs of iteration k; s_wait_asynccnt 0 + one barrier.
// ---------------------------------------------------------------------------
__global__ __launch_bounds__(256)
void conv3x3_wmma_f32_pipe(const float* __restrict__ x,
                           const float* __restrict__ wrep,
                           const float* __restrict__ bias,
                           float* __restrict__ out)
{
    __shared__ __attribute__((aligned(16))) float Ws[2][WCHUNK]; // 2 x 36 KB
    __shared__ __attribute__((aligned(16))) float Xs[2][1024];   // 2 x 4 KB

    const int tid  = threadIdx.x;
    const int tile = blockIdx.x;
    const int ow0  = (tile % 7) * 16;
    const int t2   = tile / 7;
    const int oh   = t2 % HW;
    const int bb   = t2 / HW;

    const int wave    = tid >> 5;
    const int lane    = tid & 31;
    const int half    = lane >> 4;
    const int l16     = lane & 15;
    const int co_base = wave << 4;

    const int xcol = tid & 15;
    const int ci0  = tid >> 4;

    v8f c0 = {0.f,0.f,0.f,0.f,0.f,0.f,0.f,0.f};
    v8f c1 = {0.f,0.f,0.f,0.f,0.f,0.f,0.f,0.f};

    // ---- prologue: stage kpos 0 ----
    {
        #pragma unroll
        for (int i = 0; i < 9; ++i)
            copy16_g2l(wrep + (tid << 2) + i * 1024, &Ws[0][(tid << 2) + i * 1024]);

        const int ih = oh - 1;                 // kh = 0
        const int iw = ow0 + xcol - 1;         // kw = 0
        const bool ok = ((unsigned)ih < HW) & ((unsigned)iw < HW);
        #pragma unroll
        for (int r = 0; r < 4; ++r) {
            const int ci = ci0 + (r << 4);
            float v = 0.0f;
            if (ok) v = x[(((bb * C_IN + ci) * HW) + ih) * HW + iw];
            Xs[0][(ci >> 1) * 32 + xcol * 2 + (ci & 1)] = v;
        }
        wait_async0();
        __syncthreads();
    }

    for (int kpos = 0; kpos < 9; ++kpos) {
        const int cur = kpos & 1;
        const int nxt = cur ^ 1;

        // ---- issue next iteration's staging (hidden behind the WMMAs) ----
        float xr[4];
        if (kpos < 8) {
            const float* wsrc = wrep + (kpos + 1) * WCHUNK + (tid << 2);
            #pragma unroll
            for (int i = 0; i < 9; ++i)
                copy16_g2l(wsrc + i * 1024, &Ws[nxt][(tid << 2) + i * 1024]);

            const int kh = (kpos + 1) / 3;
            const int kw = (kpos + 1) - kh * 3;
            const int ih = oh + kh - 1;
            const int iw = ow0 + xcol + kw - 1;
            const bool ok = ((unsigned)ih < HW) & ((unsigned)iw < HW);
            #pragma unroll
            for (int r = 0; r < 4; ++r) {
                const int ci = ci0 + (r << 4);
                xr[r] = 0.0f;
                if (ok) xr[r] = x[(((bb * C_IN + ci) * HW) + ih) * HW + iw];
            }
        }

        // ---- 16 WMMA k-steps on the current buffers ----
        #pragma unroll
        for (int s = 0; s < 16; ++s) {
            const int p = (s << 1) + half;     // K-pair index: kk/2 + half
            const v2f a  = *(const v2f*)&Ws[cur][p * 288 + ((co_base + l16) << 1)];
            const v2f bf = *(const v2f*)&Xs[cur][p * 32  + (l16 << 1)];
            if (s & 1)
                c1 = __builtin_amdgcn_wmma_f32_16x16x4_f32(
                        false, a, false, bf, (short)0, c1, false, false);
            else
                c0 = __builtin_amdgcn_wmma_f32_16x16x4_f32(
                        false, a, false, bf, (short)0, c0, false, false);
        }

        // ---- commit next X panel, drain async W copies, sync ----
        if (kpos < 8) {
            #pragma unroll
            for (int r = 0; r < 4; ++r) {
                const int ci = ci0 + (r << 4);
                Xs[nxt][(ci >> 1) * 32 + xcol * 2 + (ci & 1)] = xr[r];
            }
        }
        wait_async0();
        __syncthreads();
    }

    const v8f c = c0 + c1;

    #pragma unroll
    for (int j = 0; j < 8; ++j) {
        const int co = co_base + j + (half << 3);
        const float r = c[j] + bias[co];
        out[(((bb * C_OUT + co) * HW) + oh) * HW + (ow0 + l16)] = r;
    }
}

// ---------------------------------------------------------------------------
// Fallback (round-1 kernel): used if d_ws is too small for the weight repack.
// ---------------------------------------------------------------------------
__global__ __launch_bounds__(256)
void conv3x3_wmma_f32_basic(const float* __restrict__ x,
                            const float* __restrict__ w,
                            const float* __restrict__ bias,
                            float* __restrict__ out)
{
    __shared__ float Ws[WCHUNK];
    __shared__ float Xs[1024];

    const int tid  = threadIdx.x;
    const int tile = blockIdx.x;
    const int ow0  = (tile % 7) * 16;
    const int t2   = tile / 7;
    const int oh   = t2 % HW;
    const int bb   = t2 / HW;

    const int wave    = tid >> 5;
    const int lane    = tid & 31;
    const int half    = lane >> 4;
    const int l16     = lane & 15;
    const int co_base = wave << 4;

    v8f c0 = {0.f,0.f,0.f,0.f,0.f,0.f,0.f,0.f};
    v8f c1 = {0.f,0.f,0.f,0.f,0.f,0.f,0.f,0.f};

    for (int kpos = 0; kpos < 9; ++kpos) {
        const int kh = kpos / 3;
        const int kw = kpos - kh * 3;
        {
            const int ih  = oh + kh - 1;
            const int col = tid & 15;
            const int iw  = ow0 + col + kw - 1;
            const bool ok = ((unsigned)ih < HW) & ((unsigned)iw < HW);
            #pragma unroll
            for (int r = 0; r < 4; ++r) {
                const int ci = (tid >> 4) + (r << 4);
                float v = 0.0f;
                if (ok) v = x[(((bb * C_IN + ci) * HW) + ih) * HW + iw];
                Xs[(ci >> 1) * 32 + col * 2 + (ci & 1)] = v;
            }
        }
        {
            #pragma unroll 4
            for (int i = 0; i < 32; ++i) {
                const int e  = tid + (i << 8);
                const int co = e & 127;
                const int ci = e >> 7;
                Ws[(ci >> 1) * 288 + co * 2 + (ci & 1)] = w[(co * C_IN + ci) * 9 + kpos];
            }
        }
        __syncthreads();

        #pragma unroll
        for (int s = 0; s < 16; ++s) {
            const int p = (s << 1) + half;
            const v2f a  = *(const v2f*)&Ws[p * 288 + ((co_base + l16) << 1)];
            const v2f bf = *(const v2f*)&Xs[p * 32  + (l16 << 1)];
            if (s & 1)
                c1 = __builtin_amdgcn_wmma_f32_16x16x4_f32(
                        false, a, false, bf, (short)0, c1, false, false);
            else
                c0 = __builtin_amdgcn_wmma_f32_16x16x4_f32(
                        false, a, false, bf, (short)0, c0, false, false);
        }
        __syncthreads();
    }

    const v8f c = c0 + c1;
    #pragma unroll
    for (int j = 0; j < 8; ++j) {
        const int co = co_base + j + (half << 3);
        out[(((bb * C_OUT + co) * HW) + oh) * HW + (ow0 + l16)] = c[j] + bias[co];
    }
}

extern "C" void kernel_launch(void* const* d_in, const int* in_sizes, int n_in,
                              void* d_out, int out_size, void* d_ws, size_t ws_size,
                              hipStream_t stream) {
    (void)in_sizes; (void)n_in; (void)out_size;
    const float* x  = (const float*)d_in[0];
    const float* w  = (const float*)d_in[1];
    const float* b  = (const float*)d_in[2];
    float* out      = (float*)d_out;

    const int n_tiles = 16 * HW * (HW / 16);            // 12544 workgroups
    const size_t need = (size_t)9 * WCHUNK * sizeof(float);  // 324 KB

    if (d_ws != nullptr && ws_size >= need) {
        float* wrep = (float*)d_ws;
        repack_w<<<dim3(9 * 8192 / 256), dim3(256), 0, stream>>>(w, wrep);
        conv3x3_wmma_f32_pipe<<<dim3(n_tiles), dim3(256), 0, stream>>>(x, wrep, b, out);
    } else {
        conv3x3_wmma_f32_basic<<<dim3(n_tiles), dim3(256), 0, stream>>>(x, w, b, out);
    }
}